// MemoryBank_38010460570013
// MI455X (gfx1250) — compile-verified
//
#include <hip/hip_runtime.h>
#include <math.h>

// MemoryBank.get_all_distances on MI455X (gfx1250, wave32, WMMA).
//
// out[n, b] = || bank[b] - emb[n] ||_2,  emb: [64,64] f32, bank: [131072,64] f32.
// Rewritten as d^2 = ||e||^2 - 2 e.b + ||b||^2, computed with V_WMMA_F32_16X16X4_F32
// so the kernel stays on the 23.3 TB/s bandwidth floor (~64 MB traffic) instead of
// being VALU-FMA limited.

typedef __attribute__((ext_vector_type(2))) float v2f;
typedef __attribute__((ext_vector_type(4))) float v4f;
typedef __attribute__((ext_vector_type(8))) float v8f;

#define BANK_N     131072
#define DIM        64
#define BATCH      64
#define ROWSTRIDE  68            // padded LDS row stride (floats) -> conflict-free b64 reads
#define ROWS_PER_WAVE   16
#define WAVES_PER_BLOCK 8
#define ROWS_PER_BLOCK  (ROWS_PER_WAVE * WAVES_PER_BLOCK)  // 128

__global__ __launch_bounds__(256) void memorybank_dist_wmma(
    const float* __restrict__ emb,   // [64, 64]
    const float* __restrict__ bank,  // [131072, 64]
    float* __restrict__ out)         // [64, 131072]
{
    __shared__ float lds_emb[BATCH * ROWSTRIDE];                        // 17408 B
    __shared__ float lds_qn[BATCH];                                     //   256 B
    __shared__ float lds_bank[WAVES_PER_BLOCK][ROWS_PER_WAVE * ROWSTRIDE]; // 34816 B

    const int tid  = threadIdx.x;
    const int lane = tid & 31;
    const int wid  = tid >> 5;
    const int ln   = lane & 15;   // N index within tile / M index for A rows
    const int hl   = lane >> 4;   // wave half selects K sub-pair {0,1} vs {2,3}

    // ---- stage emb (64x64) into padded LDS: 4096 floats, 16 per thread ----
    #pragma unroll
    for (int i = 0; i < 4; ++i) {
        int e   = (i * 256 + tid) * 4;          // 0..4095, step 4
        int row = e >> 6, col = e & 63;
        v4f v = *(const v4f*)(emb + e);
        *(v4f*)(&lds_emb[row * ROWSTRIDE + col]) = v;
    }
    __syncthreads();

    // ---- per-query squared norms (tiny, once per block) ----
    if (tid < BATCH) {
        float s = 0.f;
        const float* r = &lds_emb[tid * ROWSTRIDE];
        #pragma unroll
        for (int d = 0; d < DIM; ++d) s += r[d] * r[d];
        lds_qn[tid] = s;
    }

    // ---- stage this wave's 16 bank rows (coalesced b128), padded ----
    const int  b0   = blockIdx.x * ROWS_PER_BLOCK + wid * ROWS_PER_WAVE;
    const float* bsrc = bank + (size_t)b0 * DIM;
    float* lb = &lds_bank[wid][0];
    #pragma unroll
    for (int i = 0; i < 8; ++i) {
        int e   = (i * 32 + lane) * 4;          // 0..1023, step 4
        int row = e >> 6, col = e & 63;
        v4f v = *(const v4f*)(bsrc + e);
        *(v4f*)(&lb[row * ROWSTRIDE + col]) = v;
    }
    __syncthreads();

    // Per-lane LDS read bases matching the 32-bit WMMA A/B layout:
    //   lanes 0-15 hold K = {4k, 4k+1}, lanes 16-31 hold K = {4k+2, 4k+3}.
    const float* bptr = &lb[ln * ROWSTRIDE + 2 * hl];
    const float* aptr = &lds_emb[ln * ROWSTRIDE + 2 * hl];

    v8f acc[4] = {v8f{}, v8f{}, v8f{}, v8f{}};
    float psq = 0.f;                             // partial bank-row sumsq (half the K range)

    #pragma unroll
    for (int kk = 0; kk < 16; ++kk) {            // K = 64 in steps of 4
        v2f b = *(const v2f*)(bptr + kk * 4);
        psq = fmaf(b.x, b.x, psq);
        psq = fmaf(b.y, b.y, psq);
        #pragma unroll
        for (int t = 0; t < 4; ++t) {            // 4 query tiles of 16
            v2f a = *(const v2f*)(aptr + t * 16 * ROWSTRIDE + kk * 4);
            acc[t] = __builtin_amdgcn_wmma_f32_16x16x4_f32(
                /*neg_a=*/false, a, /*neg_b=*/false, b,
                /*c_mod=*/(short)0, acc[t], /*reuse_a=*/false, /*reuse_b=*/false);
        }
    }

    // full ||bank_row||^2: lane L and L^16 hold complementary K halves of row (L&15)
    float bn = psq + __shfl_xor(psq, 16, 32);

    // ---- epilogue: d = sqrt(max(0, qn + bn - 2*dot)), coalesced b32 stores ----
    #pragma unroll
    for (int t = 0; t < 4; ++t) {
        #pragma unroll
        for (int r = 0; r < 8; ++r) {
            int m = t * 16 + r + 8 * hl;         // C/D layout: VGPR r -> M = r (+8 for hi half)
            float d2 = lds_qn[m] + bn - 2.0f * acc[t][r];
            d2 = fmaxf(d2, 0.0f);
            out[(size_t)m * BANK_N + b0 + ln] = __builtin_sqrtf(d2);
        }
    }
}

extern "C" void kernel_launch(void* const* d_in, const int* in_sizes, int n_in,
                              void* d_out, int out_size, void* d_ws, size_t ws_size,
                              hipStream_t stream) {
    (void)in_sizes; (void)n_in; (void)d_ws; (void)ws_size; (void)out_size;
    const float* emb  = (const float*)d_in[0];
    const float* bank = (const float*)d_in[1];
    float* out = (float*)d_out;

    dim3 grid(BANK_N / ROWS_PER_BLOCK);   // 1024 blocks
    dim3 block(256);                      // 8 wave32 per block
    hipLaunchKernelGGL(memorybank_dist_wmma, grid, block, 0, stream, emb, bank, out);
}